// ABCNN_3_58188216926320
// MI455X (gfx1250) — compile-verified
//
#include <hip/hip_runtime.h>
#include <hip/hip_bf16.h>

#define Bn   64
#define Ln   512
#define Dn   512
#define Wn   4
#define LOn  515   // L + W - 1 (conv output length)
#define LPn  576   // LOn padded up to 9*64 for clean 64x64 GEMM tiling

typedef __attribute__((ext_vector_type(2))) float v2f;
typedef __attribute__((ext_vector_type(8))) float v8f;

// ---------------------------------------------------------------------------
// gfx1250 async global->LDS copy (VGLOBAL GLOBAL_LOAD_ASYNC_TO_LDS_B128,
// tracked by ASYNCcnt).  lds_off = low 32 bits of a generic pointer to
// __shared__ (the flat LDS aperture keeps the workgroup-relative offset in
// addr[31:0], which is exactly what the instruction's VDST operand wants).
// ---------------------------------------------------------------------------
__device__ __forceinline__ void async_copy_b128(unsigned lds_off, const float* g) {
    asm volatile("global_load_async_to_lds_b128 %0, %1, off"
                 :: "v"(lds_off), "v"((unsigned long long)(uintptr_t)g)
                 : "memory");
}
__device__ __forceinline__ void wait_async0() {
    asm volatile("s_wait_asynccnt 0x0" ::: "memory");
}
__device__ __forceinline__ unsigned lds_off_of(const void* p) {
    return (unsigned)(uintptr_t)p;
}

// ---------------------------------------------------------------------------
// K1: F = Fr * mask (broadcast over D), plus per-row squared norm.
// grid (L, B), block 128
// ---------------------------------------------------------------------------
__global__ void mask_sqnorm_kernel(const float* __restrict__ Fr,
                                   const float* __restrict__ mask,
                                   float* __restrict__ Fm,
                                   float* __restrict__ sqn) {
    const int i = blockIdx.x, b = blockIdx.y, tid = threadIdx.x;
    const float m = mask[b * Ln + i];
    const float* src = Fr + ((size_t)b * Ln + i) * Dn;
    float*       dst = Fm + ((size_t)b * Ln + i) * Dn;
    float s = 0.f;
    for (int d = tid; d < Dn; d += 128) {
        float v = src[d] * m;
        dst[d] = v;
        s += v * v;
    }
    __shared__ float red[128];
    red[tid] = s; __syncthreads();
    for (int o = 64; o > 0; o >>= 1) {
        if (tid < o) red[tid] += red[tid + o];
        __syncthreads();
    }
    if (tid == 0) sqn[b * Ln + i] = red[0];
}

// ---------------------------------------------------------------------------
// K2: distance-similarity GEMM.  Aout[b,i,j] = 1/(1+sqrt(max(|x_i|^2+|y_j|^2
//     - 2 x_i.y_j, 0))) for i<Mreal, j<Nreal; 0 in the padded region.
// 128 threads / 4 waves, 64x64 block tile, each wave owns a 32x32 sub-tile
// as 2x2 accumulators (2 A-frags x 2 B-frags -> 4 independent WMMA chains,
// 1.0 LDS fragment loads per wmma).  K chunks of 32, double-buffered LDS
// filled by async global->LDS copies; ONE barrier per chunk.
// ---------------------------------------------------------------------------
__global__ __launch_bounds__(128)
void dist_gemm_kernel(const float* __restrict__ X, const float* __restrict__ Y,
                      const float* __restrict__ sqx, const float* __restrict__ sqy,
                      float* __restrict__ Aout,
                      int Mreal, int Nreal, int ldOut,
                      size_t sX, size_t sY, size_t sOut, int ssx, int ssy) {
    const int b  = blockIdx.z;
    const int i0 = blockIdx.y * 64;
    const int j0 = blockIdx.x * 64;
    const int t  = threadIdx.x;
    const int l  = t & 31;
    const int half = l >> 4, lr = l & 15;
    const int w  = t >> 5;                 // 0..3
    const int mW = (w & 1) * 32;
    const int nW = (w >> 1) * 32;

    __shared__ __align__(16) float Xs[2][64][36];
    __shared__ __align__(16) float Ys[2][64][36];

    const float* Xb = X + sX * b;
    const float* Yb = Y + sY * b;

    // cooperative tile: 64x32 floats per matrix, 128 threads -> 4 b128 each
    const int r0   = t >> 3;               // 0..15, rows r0 + 16q
    const int lcol = (t & 7) * 4;
    const float* gx = Xb + (size_t)(i0 + r0) * Dn + lcol;
    const float* gy = Yb + (size_t)(j0 + r0) * Dn + lcol;
    unsigned xo[2], yo[2];
    xo[0] = lds_off_of(&Xs[0][r0][lcol]); xo[1] = lds_off_of(&Xs[1][r0][lcol]);
    yo[0] = lds_off_of(&Ys[0][r0][lcol]); yo[1] = lds_off_of(&Ys[1][r0][lcol]);
    const unsigned span16 = 16u * 36u * 4u;   // +16 rows in LDS bytes
    const size_t   gstep16 = (size_t)16 * Dn; // +16 rows in elements

    auto issue = [&](int buf, int k0) {
        #pragma unroll
        for (int q = 0; q < 4; ++q) {
            async_copy_b128(xo[buf] + q * span16, gx + k0 + q * gstep16);
            async_copy_b128(yo[buf] + q * span16, gy + k0 + q * gstep16);
        }
    };

    v8f acc00 = {}, acc01 = {}, acc10 = {}, acc11 = {};
    issue(0, 0);
    for (int k0 = 0; k0 < Dn; k0 += 32) {
        const int cur = (k0 >> 5) & 1;
        wait_async0();
        __syncthreads();
        if (k0 + 32 < Dn) issue(cur ^ 1, k0 + 32);
        #pragma unroll
        for (int kk = 0; kk < 32; kk += 4) {
            const int kf = kk + 2 * half;
            v2f a0, a1, b0, b1;
            a0.x = Xs[cur][mW + lr][kf];      a0.y = Xs[cur][mW + lr][kf + 1];
            a1.x = Xs[cur][mW + 16 + lr][kf]; a1.y = Xs[cur][mW + 16 + lr][kf + 1];
            b0.x = Ys[cur][nW + lr][kf];      b0.y = Ys[cur][nW + lr][kf + 1];
            b1.x = Ys[cur][nW + 16 + lr][kf]; b1.y = Ys[cur][nW + 16 + lr][kf + 1];
            acc00 = __builtin_amdgcn_wmma_f32_16x16x4_f32(
                false, a0, false, b0, (short)0, acc00, false, false);
            acc01 = __builtin_amdgcn_wmma_f32_16x16x4_f32(
                false, a0, false, b1, (short)0, acc01, false, false);
            acc10 = __builtin_amdgcn_wmma_f32_16x16x4_f32(
                false, a1, false, b0, (short)0, acc10, false, false);
            acc11 = __builtin_amdgcn_wmma_f32_16x16x4_f32(
                false, a1, false, b1, (short)0, acc11, false, false);
        }
    }

    // epilogue: fused 1/(1+dist) with padding mask
    const int ja = j0 + nW + lr;
    const int jb = ja + 16;
    const float sya = sqy[(size_t)b * ssy + ja];
    const float syb = sqy[(size_t)b * ssy + jb];
    float* Ob = Aout + sOut * b;
    #pragma unroll
    for (int r = 0; r < 8; ++r) {
        const int ia = i0 + mW + r + 8 * half;
        const int ib = ia + 16;
        const float sxa = sqx[(size_t)b * ssx + ia];
        const float sxb = sqx[(size_t)b * ssx + ib];
        const float v00 = 1.f / (1.f + sqrtf(fmaxf(sxa + sya - 2.f * acc00[r], 0.f)));
        const float v01 = 1.f / (1.f + sqrtf(fmaxf(sxa + syb - 2.f * acc01[r], 0.f)));
        const float v10 = 1.f / (1.f + sqrtf(fmaxf(sxb + sya - 2.f * acc10[r], 0.f)));
        const float v11 = 1.f / (1.f + sqrtf(fmaxf(sxb + syb - 2.f * acc11[r], 0.f)));
        Ob[(size_t)ia * ldOut + ja] = (ia < Mreal && ja < Nreal) ? v00 : 0.f;
        Ob[(size_t)ia * ldOut + jb] = (ia < Mreal && jb < Nreal) ? v01 : 0.f;
        Ob[(size_t)ib * ldOut + ja] = (ib < Mreal && ja < Nreal) ? v10 : 0.f;
        Ob[(size_t)ib * ldOut + jb] = (ib < Mreal && jb < Nreal) ? v11 : 0.f;
    }
}

// ---------------------------------------------------------------------------
// K3: attention feature GEMM.  C[b] = op(A[b]) * Wm   (all 512x512 row-major)
//   TRANSA=0: C[i,d] = sum_j A[i,j] W[j,d]   (F1a); X tile stored [m][k]
//   TRANSA=1: C[j,d] = sum_i A[i,j] W[i,d]   (F0a); X tile stored copy-through
//             as [k][m] and the A-fragment is read with B-style indexing --
//             the transpose is free, so async copies work for both cases.
// Same 128-thread / 4-wave / 2x2-accumulator structure as dist_gemm.
// ---------------------------------------------------------------------------
template <int TRANSA>
__global__ __launch_bounds__(128)
void attn_gemm_kernel(const float* __restrict__ Amat, const float* __restrict__ Wm,
                      float* __restrict__ Cout) {
    constexpr int XR = TRANSA ? 32 : 64;
    constexpr int XP = TRANSA ? 68 : 36;
    __shared__ __align__(16) float Xt[2][XR][XP];
    __shared__ __align__(16) float Wt[2][32][68];

    const int b  = blockIdx.z;
    const int i0 = blockIdx.y * 64;   // output rows
    const int j0 = blockIdx.x * 64;   // output cols (d)
    const int t  = threadIdx.x;
    const int l  = t & 31;
    const int half = l >> 4, lr = l & 15;
    const int w  = t >> 5;
    const int mW = (w & 1) * 32;
    const int nW = (w >> 1) * 32;

    const float* Ab = Amat + (size_t)b * Ln * Ln;

    // X-tile cooperative load coords (2048 floats -> 4 b128 per thread)
    int xr0, xcol, xrstep;
    if (TRANSA) { xr0 = t >> 4; xcol = (t & 15) * 4; xrstep = 8;  }  // [32 k][64 m]
    else        { xr0 = t >> 3; xcol = (t & 7) * 4;  xrstep = 16; }  // [64 m][32 k]
    // W-tile: [32 k][64 n], rows wr0 + 8q
    const int wr0 = t >> 4;
    const int wcol = (t & 15) * 4;

    unsigned xoff[2], woff[2];
    #pragma unroll
    for (int q = 0; q < 2; ++q) {
        xoff[q] = lds_off_of(&Xt[q][xr0][xcol]);
        woff[q] = lds_off_of(&Wt[q][wr0][wcol]);
    }
    const unsigned xspan = (unsigned)xrstep * XP * 4u;
    const unsigned wspan = 8u * 68u * 4u;

    auto issue = [&](int buf, int k0) {
        #pragma unroll
        for (int q = 0; q < 4; ++q) {
            const float* p = TRANSA
                ? Ab + (size_t)(k0 + xr0 + q * xrstep) * Ln + i0 + xcol
                : Ab + (size_t)(i0 + xr0 + q * xrstep) * Ln + k0 + xcol;
            async_copy_b128(xoff[buf] + q * xspan, p);
            async_copy_b128(woff[buf] + q * wspan,
                            Wm + (size_t)(k0 + wr0 + q * 8) * Dn + j0 + wcol);
        }
    };

    v8f acc00 = {}, acc01 = {}, acc10 = {}, acc11 = {};
    issue(0, 0);
    for (int k0 = 0; k0 < Ln; k0 += 32) {
        const int cur = (k0 >> 5) & 1;
        wait_async0();
        __syncthreads();
        if (k0 + 32 < Ln) issue(cur ^ 1, k0 + 32);
        #pragma unroll
        for (int kk = 0; kk < 32; kk += 4) {
            const int kf = kk + 2 * half;
            v2f a0, a1, b0, b1;
            if (TRANSA) {
                a0.x = Xt[cur][kf][mW + lr];      a0.y = Xt[cur][kf + 1][mW + lr];
                a1.x = Xt[cur][kf][mW + 16 + lr]; a1.y = Xt[cur][kf + 1][mW + 16 + lr];
            } else {
                a0.x = Xt[cur][mW + lr][kf];      a0.y = Xt[cur][mW + lr][kf + 1];
                a1.x = Xt[cur][mW + 16 + lr][kf]; a1.y = Xt[cur][mW + 16 + lr][kf + 1];
            }
            b0.x = Wt[cur][kf][nW + lr];      b0.y = Wt[cur][kf + 1][nW + lr];
            b1.x = Wt[cur][kf][nW + 16 + lr]; b1.y = Wt[cur][kf + 1][nW + 16 + lr];
            acc00 = __builtin_amdgcn_wmma_f32_16x16x4_f32(
                false, a0, false, b0, (short)0, acc00, false, false);
            acc01 = __builtin_amdgcn_wmma_f32_16x16x4_f32(
                false, a0, false, b1, (short)0, acc01, false, false);
            acc10 = __builtin_amdgcn_wmma_f32_16x16x4_f32(
                false, a1, false, b0, (short)0, acc10, false, false);
            acc11 = __builtin_amdgcn_wmma_f32_16x16x4_f32(
                false, a1, false, b1, (short)0, acc11, false, false);
        }
    }

    float* Cb = Cout + (size_t)b * Ln * Dn;
    const int ja = j0 + nW + lr;
    #pragma unroll
    for (int r = 0; r < 8; ++r) {
        const int ia = i0 + mW + r + 8 * half;
        const int ib = ia + 16;
        Cb[(size_t)ia * Dn + ja]      = acc00[r];
        Cb[(size_t)ia * Dn + ja + 16] = acc01[r];
        Cb[(size_t)ib * Dn + ja]      = acc10[r];
        Cb[(size_t)ib * Dn + ja + 16] = acc11[r];
    }
}

// ---------------------------------------------------------------------------
// K4: width-4 conv over L (2 input channels), +bias, tanh; writes padded
// [B, LPn, D] output (zero rows >= LOn) and per-row squared norms.
// grid (LPn, B), block 256
// ---------------------------------------------------------------------------
__global__ void conv_tanh_kernel(const float* __restrict__ c0,
                                 const float* __restrict__ c1,
                                 const float* __restrict__ cw,
                                 const float* __restrict__ cb,
                                 float* __restrict__ out,
                                 float* __restrict__ sqo) {
    const int tpos = blockIdx.x, b = blockIdx.y, tid = threadIdx.x;
    float* orow = out + ((size_t)b * LPn + tpos) * Dn;
    float s = 0.f;
    if (tpos < LOn) {
        float wk[8];
        #pragma unroll
        for (int q = 0; q < 8; ++q) wk[q] = cw[q];   // [c][k] = cw[c*4+k]
        const float bias = cb[0];
        const size_t base = (size_t)b * Ln * Dn;
        for (int d = tid; d < Dn; d += 256) {
            float a = bias;
            #pragma unroll
            for (int k = 0; k < Wn; ++k) {
                const int src = tpos + k - (Wn - 1);
                if (src >= 0 && src < Ln) {
                    const size_t off = base + (size_t)src * Dn + d;
                    a += wk[k] * c0[off] + wk[4 + k] * c1[off];
                }
            }
            const float v = tanhf(a);
            orow[d] = v;
            s += v * v;
        }
    } else {
        for (int d = tid; d < Dn; d += 256) orow[d] = 0.f;
    }
    __shared__ float red[256];
    red[tid] = s; __syncthreads();
    for (int o = 128; o > 0; o >>= 1) {
        if (tid < o) red[tid] += red[tid + o];
        __syncthreads();
    }
    if (tid == 0) sqo[b * LPn + tpos] = (tpos < LOn) ? red[0] : 0.f;
}

// ---------------------------------------------------------------------------
// K5a: row sums of A2 (over j < LOn).  grid (LOn, B), block 256
// ---------------------------------------------------------------------------
__global__ void rowsum_kernel(const float* __restrict__ A2, float* __restrict__ rs) {
    const int i = blockIdx.x, b = blockIdx.y, tid = threadIdx.x;
    const float* row = A2 + ((size_t)b * LPn + i) * LPn;
    float s = 0.f;
    for (int j = tid; j < LOn; j += 256) s += row[j];
    __shared__ float red[256];
    red[tid] = s; __syncthreads();
    for (int o = 128; o > 0; o >>= 1) {
        if (tid < o) red[tid] += red[tid + o];
        __syncthreads();
    }
    if (tid == 0) rs[b * LPn + i] = red[0];
}

// ---------------------------------------------------------------------------
// K5b: col sums of A2 (over i < LOn).  grid (ceil(LOn/256), B), block 256
// ---------------------------------------------------------------------------
__global__ void colsum_kernel(const float* __restrict__ A2, float* __restrict__ cs) {
    const int j = blockIdx.x * 256 + threadIdx.x;
    const int b = blockIdx.y;
    if (j >= LOn) return;
    const float* base = A2 + (size_t)b * LPn * LPn + j;
    float s = 0.f;
    for (int i = 0; i < LOn; ++i) s += base[(size_t)i * LPn];
    cs[b * LPn + j] = s;
}

// ---------------------------------------------------------------------------
// K6: weighted avg-pool: o[b,t,d] = 0.25 * sum_k out[b,t+k,d] * w[b,t+k]
// grid (L, B), block 256
// ---------------------------------------------------------------------------
__global__ void pool_kernel(const float* __restrict__ ox,
                            const float* __restrict__ wsum,
                            float* __restrict__ o) {
    const int tpos = blockIdx.x, b = blockIdx.y;
    float sc[Wn];
    #pragma unroll
    for (int k = 0; k < Wn; ++k) sc[k] = wsum[b * LPn + tpos + k];
    const float* base = ox + ((size_t)b * LPn + tpos) * Dn;
    float* dst = o + ((size_t)b * Ln + tpos) * Dn;
    for (int d = threadIdx.x; d < Dn; d += 256) {
        float s = 0.f;
        #pragma unroll
        for (int k = 0; k < Wn; ++k) s += base[(size_t)k * Dn + d] * sc[k];
        dst[d] = 0.25f * s;
    }
}

// ---------------------------------------------------------------------------
extern "C" void kernel_launch(void* const* d_in, const int* in_sizes, int n_in,
                              void* d_out, int out_size, void* d_ws, size_t ws_size,
                              hipStream_t stream) {
    const float* F0r = (const float*)d_in[0];
    const float* F1r = (const float*)d_in[1];
    const float* m0  = (const float*)d_in[2];
    const float* m1  = (const float*)d_in[3];
    const float* W0  = (const float*)d_in[4];
    const float* W1  = (const float*)d_in[5];
    const float* cw  = (const float*)d_in[6];
    const float* cb  = (const float*)d_in[7];

    float* ws = (float*)d_ws;
    const size_t nBLD  = (size_t)Bn * Ln * Dn;      // 16.78M
    const size_t nBLL  = (size_t)Bn * Ln * Ln;      // 16.78M
    const size_t nBLPD = (size_t)Bn * LPn * Dn;     // 18.87M

    float* F0   = ws;                                // [B,L,D]
    float* F1   = F0 + nBLD;                         // [B,L,D]
    float* Amat = F1 + nBLD;                         // [B,L,L]
    float* F0a  = Amat + nBLL;                       // [B,L,D]
    float* F1a  = F0a + nBLD;                        // [B,L,D]
    float* out0 = F1a + nBLD;                        // [B,LP,D] (rows>=LO zero)
    float* out1 = out0 + nBLPD;                      // [B,LP,D]
    float* sqn0 = out1 + nBLPD;                      // [B,L]
    float* sqn1 = sqn0 + (size_t)Bn * Ln;            // [B,L]
    float* sqo0 = sqn1 + (size_t)Bn * Ln;            // [B,LP]
    float* sqo1 = sqo0 + (size_t)Bn * LPn;           // [B,LP]
    float* rs   = sqo1 + (size_t)Bn * LPn;           // [B,LP]
    float* cs   = rs   + (size_t)Bn * LPn;           // [B,LP]
    // A2 [B,LP,LP] (21.2M) aliases F0+F1 (33.55M): F0/F1 are dead once the
    // conv kernels have run, and A2 is produced strictly after them.
    float* A2 = F0;

    // 1) mask + squared row norms
    mask_sqnorm_kernel<<<dim3(Ln, Bn), 128, 0, stream>>>(F0r, m0, F0, sqn0);
    mask_sqnorm_kernel<<<dim3(Ln, Bn), 128, 0, stream>>>(F1r, m1, F1, sqn1);

    // 2) A = 1/(1+pdist(F0,F1)) : batched 512x512x512 WMMA GEMM + epilogue
    dist_gemm_kernel<<<dim3(8, 8, Bn), 128, 0, stream>>>(
        F0, F1, sqn0, sqn1, Amat, Ln, Ln, Ln,
        (size_t)Ln * Dn, (size_t)Ln * Dn, (size_t)Ln * Ln, Ln, Ln);

    // 3) F0a = A^T W0 ; F1a = A W1
    attn_gemm_kernel<1><<<dim3(8, 8, Bn), 128, 0, stream>>>(Amat, W0, F0a);
    attn_gemm_kernel<0><<<dim3(8, 8, Bn), 128, 0, stream>>>(Amat, W1, F1a);

    // 4) conv + tanh (+ padded rows + row sqnorms)
    conv_tanh_kernel<<<dim3(LPn, Bn), 256, 0, stream>>>(F0, F0a, cw, cb, out0, sqo0);
    conv_tanh_kernel<<<dim3(LPn, Bn), 256, 0, stream>>>(F1, F1a, cw, cb, out1, sqo1);

    // 5) A2 = 1/(1+pdist(out0,out1)) on padded 576x576 grid, masked to 515
    dist_gemm_kernel<<<dim3(9, 9, Bn), 128, 0, stream>>>(
        out0, out1, sqo0, sqo1, A2, LOn, LOn, LPn,
        (size_t)LPn * Dn, (size_t)LPn * Dn, (size_t)LPn * LPn, LPn, LPn);

    // 6) attention weights: row sums (for out0), col sums (for out1)
    rowsum_kernel<<<dim3(LOn, Bn), 256, 0, stream>>>(A2, rs);
    colsum_kernel<<<dim3((LOn + 255) / 256, Bn), 256, 0, stream>>>(A2, cs);

    // 7) weighted avg-pool -> (o0, o1) concatenated in d_out
    float* o0 = (float*)d_out;
    float* o1 = o0 + nBLD;
    pool_kernel<<<dim3(Ln, Bn), 256, 0, stream>>>(out0, rs, o0);
    pool_kernel<<<dim3(Ln, Bn), 256, 0, stream>>>(out1, cs, o1);
}